// Alignment_Generator_89558658056847
// MI455X (gfx1250) — compile-verified
//
#include <hip/hip_runtime.h>

// Alignment generator (length regulator): out[b, f, t] = 1.0 iff
// ends[b,t-1] <= f < ends[b,t] with ends = inclusive cumsum of durations.
// Store-bandwidth bound: ~260 MB written once -> floor ~11us @ 23.3 TB/s.

#define TOK 512   // tokens per row (T)
#define FPB 8     // frames per block = waves per block (wave32)

typedef float        v4f __attribute__((ext_vector_type(4)));
typedef unsigned int u32;
typedef u32          v4u __attribute__((ext_vector_type(4)));
typedef int          v8i __attribute__((ext_vector_type(8)));
typedef int          v4i __attribute__((ext_vector_type(4)));

#if __has_builtin(__builtin_amdgcn_tensor_load_to_lds) && __has_builtin(__builtin_amdgcn_s_wait_tensorcnt)
#define USE_TDM 1
#else
#define USE_TDM 0
#endif

__global__ __launch_bounds__(FPB * 32) void align_kernel(const int* __restrict__ dur,
                                                         float* __restrict__ out,
                                                         int M)
{
    // s_mem[0..511]   : duration row (TDM landing zone; LDS offset 0 — sole LDS alloc)
    // s_mem[512..1023]: inclusive cumsum (ends)
    __shared__ int s_mem[2 * TOK];
    int* s_dur  = s_mem;
    int* s_ends = s_mem + TOK;

    const int b    = blockIdx.y;
    const int tid  = (int)threadIdx.x;
    const int wave = tid >> 5;
    const int lane = tid & 31;
    const int* row = dur + b * TOK;

    if (wave == 0) {
        int v[16];
        int sum = 0;
#if USE_TDM
        // --- Tensor Data Mover: DMA one 512-int row (2KB) global -> LDS ---
        const unsigned long long ga = (unsigned long long)(const void*)row;
        v4u g0;
        g0.x = 1u;                                            // count=1, is_restore=0, gather=0
        g0.y = 0u;                                            // lds_addr = 0 (s_mem base)
        g0.z = (u32)(ga & 0xFFFFFFFFu);                       // global_addr[31:0]
        g0.w = (u32)((ga >> 32) & 0x01FFFFFFu) | (2u << 30);  // global_addr[56:32], type=2
        v8i g1;
        g1[0] = (2 << 16);                                    // wg_mask=0, data_size=2 (4B)
        g1[1] = (int)((u32)(TOK & 0xFFFF) << 16);             // tensor_dim0[15:0] @ bits[63:48]
        g1[2] = (int)(((u32)TOK >> 16) | (1u << 16));         // tensor_dim0[31:16], tensor_dim1=1
        g1[3] = (int)((u32)(TOK & 0xFFFF) << 16);             // tensor_dim1 hi=0, tile_dim0=512
        g1[4] = 1;                                            // tile_dim1=1, tile_dim2=0
        g1[5] = TOK;                                          // tensor_dim0_stride[31:0]
        g1[6] = 0;                                            // stride0 hi16=0, stride1 lo16=0
        g1[7] = 0;                                            // stride1 hi32=0
        v4i gz4 = {0, 0, 0, 0};                               // groups 2/3: unused (<=2D tensor)
        v8i gz8 = {0, 0, 0, 0, 0, 0, 0, 0};                   // trailing group (clang-23 form)
        __builtin_amdgcn_tensor_load_to_lds(g0, g1, gz4, gz4, gz8, 0);
        __builtin_amdgcn_s_wait_tensorcnt(0);                 // s_wait_tensorcnt 0
        #pragma unroll
        for (int k = 0; k < 16; ++k) { v[k] = s_dur[lane * 16 + k]; sum += v[k]; }
#else
        #pragma unroll
        for (int k = 0; k < 16; ++k) { v[k] = row[lane * 16 + k]; sum += v[k]; }
#endif
        // wave32 inclusive scan of per-lane partial sums (warpSize == 32 on gfx1250)
        int scan = sum;
        #pragma unroll
        for (int d = 1; d < 32; d <<= 1) {
            int n = __shfl_up(scan, (unsigned)d, 32);
            if (lane >= d) scan += n;
        }
        int acc = scan - sum;  // exclusive prefix of this lane's 16-element chunk
        #pragma unroll
        for (int k = 0; k < 16; ++k) { acc += v[k]; s_ends[lane * 16 + k] = acc; }
    }
    __syncthreads();

    const int f = blockIdx.x * FPB + wave;   // one frame per wave
    if (f >= M) return;

    const int total = s_ends[TOK - 1];
    int j = -1;
    if (f < total) {
        // branchless lower-bound: count of ends[] <= f  (broadcast LDS reads)
        int lo = 0;
        #pragma unroll
        for (int step = TOK / 2; step > 0; step >>= 1) {
            const int probe = lo + step;
            if (probe <= TOK && s_ends[probe - 1] <= f) lo = probe;
        }
        j = lo;  // unique token: ends[j-1] <= f < ends[j]
    }

    // Stream the 512-float row: 128 x float4, coalesced, non-temporal
    // (output >> 192MB L2, written once, never re-read by the kernel).
    v4f* orow = (v4f*)(out + ((size_t)b * (size_t)M + (size_t)f) * TOK);
    #pragma unroll
    for (int k = 0; k < 4; ++k) {
        const int i  = lane + 32 * k;   // consecutive lanes -> consecutive 16B
        const int t0 = i * 4;
        v4f val;
        val.x = (j == t0    ) ? 1.0f : 0.0f;
        val.y = (j == t0 + 1) ? 1.0f : 0.0f;
        val.z = (j == t0 + 2) ? 1.0f : 0.0f;
        val.w = (j == t0 + 3) ? 1.0f : 0.0f;
        __builtin_nontemporal_store(val, orow + i);
    }
}

extern "C" void kernel_launch(void* const* d_in, const int* in_sizes, int n_in,
                              void* d_out, int out_size, void* d_ws, size_t ws_size,
                              hipStream_t stream) {
    (void)n_in; (void)d_ws; (void)ws_size;
    const int* dur = (const int*)d_in[0];
    const int BT = in_sizes[0];        // B*T (= 32*512 per reference)
    const int Bn = BT / TOK;           // batch rows
    const int M  = out_size / BT;      // max_frames, derived host-side (no device read)
    if (M <= 0 || Bn <= 0) return;
    dim3 block(FPB * 32);
    dim3 grid((unsigned)((M + FPB - 1) / FPB), (unsigned)Bn, 1);
    align_kernel<<<grid, block, 0, stream>>>(dur, (float*)d_out, M);
}